// WindowAttention_65171833750201
// MI455X (gfx1250) — compile-verified
//
#include <hip/hip_runtime.h>
#include <hip/hip_bf16.h>

#define DIM      192
#define HEADS    6
#define HD       56
#define N_TOK    49
#define NP       64      // padded token count (4 x 16 tiles)
#define NWIN     64
#define QKVW     1008    // qkv output width
#define HSL      168     // per-head qkv slice (q|k|v = 56*3)
#define HSLP     176     // padded LDS stride for qkv head slice
#define SCALE_Q  0.17677669529663687f   // (192/6)^-0.5 = 1/sqrt(32)

typedef __attribute__((ext_vector_type(2))) float v2f;
typedef __attribute__((ext_vector_type(4))) float v4f;
typedef __attribute__((ext_vector_type(8))) float v8f;

__device__ __forceinline__ v8f wmma_f32(v2f a, v2f b, v8f c) {
  // V_WMMA_F32_16X16X4_F32: D(16x16 f32) = A(16x4) * B(4x16) + C
  return __builtin_amdgcn_wmma_f32_16x16x4_f32(
      /*neg_a=*/false, a, /*neg_b=*/false, b,
      /*c_mod=*/(short)0, c, /*reuse_a=*/false, /*reuse_b=*/false);
}

__global__ __launch_bounds__(256, 2)
void cswin_window_attn_kernel(const float* __restrict__ x,        // (2048,49,192)
                              const float* __restrict__ mask,     // (64,49,49)
                              const float* __restrict__ qkv_w,    // (192,1008)
                              const float* __restrict__ qkv_b,    // (1008)
                              const float* __restrict__ bias_tab, // (169,6)
                              const int*   __restrict__ rel_index,// (49,49)
                              const float* __restrict__ lepe_w,   // (336,1,3,3)
                              const float* __restrict__ lepe_b,   // (336)
                              const float* __restrict__ proj_w,   // (336,192)
                              const float* __restrict__ proj_b,   // (192)
                              float* __restrict__ out)            // (2048,49,192)
{
  __shared__ float xs[NP * DIM];     // x tile, rows 49..63 zero      (49 KB)
  __shared__ float qkvs[NP * HSLP];  // per-head q|k|v                (44 KB)
  __shared__ float attns[NP * NP];   // attention logits / probs      (16 KB)
  __shared__ float oh[NP * NP];      // per-head output (+lepe)       (16 KB)

  const int b     = blockIdx.x;
  const int tid   = threadIdx.x;
  const int lane  = tid & 31;
  const int wave  = tid >> 5;
  const int laneM = lane & 15;   // N index / M index within tile
  const int laneH = lane >> 4;   // half-wave selector
  const int wmask = b & (NWIN - 1);

  // ---- stage x[b] (49x192) into LDS, zero-pad rows to 64, float4 wide ----
  {
    const v4f* xg = (const v4f*)(x + (size_t)b * N_TOK * DIM);
    v4f* xl = (v4f*)xs;
    for (int idx = tid; idx < NP * (DIM / 4); idx += 256) {
      int r = idx / (DIM / 4);
      v4f val = {0.f, 0.f, 0.f, 0.f};
      if (r < N_TOK) val = xg[idx];
      xl[idx] = val;
    }
  }

  // final 49x192 accumulator: wave owns mt = wave/2, nt = (wave&1)*6 + j
  v8f facc[6];
  #pragma unroll
  for (int j = 0; j < 6; ++j) facc[j] = (v8f){};
  const int fmt  = wave >> 1;
  const int fntb = (wave & 1) * 6;

  __syncthreads();

  for (int h = 0; h < HEADS; ++h) {
    // ===== GEMM1: qkv_head = xs @ qkv_w[:, h*168 : +168] + bias ==========
    // Each wave owns one N-strip (nt) and all 4 M-tiles: B fragment is
    // fetched from global once per K-step and reused by 4 WMMAs.
    for (int nt = wave; nt < 11; nt += 8) {      // wave-uniform
      v8f acc[4];
      #pragma unroll
      for (int mt = 0; mt < 4; ++mt) acc[mt] = (v8f){};
      const int   n    = nt * 16 + laneM;        // 0..175 (>=168 is pad)
      const int   nc   = (n < HSL) ? n : (HSL - 1);
      const float nval = (n < HSL) ? 1.f : 0.f;
      const float* wcol = qkv_w + h * HSL + nc;
      for (int kk = 0; kk < DIM / 4; ++kk) {
        const int k0 = kk * 4 + laneH * 2;
        v2f bb;
        bb.x = wcol[(size_t)k0 * QKVW] * nval;
        bb.y = wcol[(size_t)(k0 + 1) * QKVW] * nval;
        #pragma unroll
        for (int mt = 0; mt < 4; ++mt) {
          v2f a;
          a.x = xs[(mt * 16 + laneM) * DIM + k0];
          a.y = xs[(mt * 16 + laneM) * DIM + k0 + 1];
          acc[mt] = wmma_f32(a, bb, acc[mt]);
        }
      }
      const float bias = (n < HSL) ? qkv_b[h * HSL + nc] : 0.f;
      #pragma unroll
      for (int mt = 0; mt < 4; ++mt) {
        #pragma unroll
        for (int i = 0; i < 8; ++i) {
          const int row = mt * 16 + i + laneH * 8;
          qkvs[row * HSLP + n] = acc[mt][i] + bias;
        }
      }
    }
    __syncthreads();

    // ===== GEMM2: attn = (q * scale) @ k^T + relbias + mask ==============
    for (int t = wave; t < 16; t += 8) {
      const int mt = t >> 2, nt = t & 3;
      v8f acc = (v8f){};
      const float* qrow = qkvs + (mt * 16 + laneM) * HSLP;       // q
      const float* kcol = qkvs + (nt * 16 + laneM) * HSLP + HD;  // k (B = k^T)
      for (int kk = 0; kk < HD / 4; ++kk) {
        const int k0 = kk * 4 + laneH * 2;
        v2f a, bb;
        a.x  = qrow[k0] * SCALE_Q;
        a.y  = qrow[k0 + 1] * SCALE_Q;
        bb.x = kcol[k0];
        bb.y = kcol[k0 + 1];
        acc = wmma_f32(a, bb, acc);
      }
      const int mTok = nt * 16 + laneM;
      #pragma unroll
      for (int i = 0; i < 8; ++i) {
        const int nTok = mt * 16 + i + laneH * 8;
        float v = acc[i];
        if (nTok < N_TOK && mTok < N_TOK) {
          const int ri = rel_index[nTok * N_TOK + mTok];
          v += bias_tab[ri * HEADS + h] +
               mask[(size_t)wmask * N_TOK * N_TOK + nTok * N_TOK + mTok];
        }
        attns[nTok * NP + mTok] = v;
      }
    }
    __syncthreads();

    // ===== softmax over valid 49 columns, zero padding ===================
    if (tid < NP) {
      float* arow = attns + tid * NP;
      if (tid < N_TOK) {
        float mx = -1e30f;
        for (int c = 0; c < N_TOK; ++c) mx = fmaxf(mx, arow[c]);
        float s = 0.f;
        for (int c = 0; c < N_TOK; ++c) {
          const float e = __expf(arow[c] - mx);
          arow[c] = e;
          s += e;
        }
        const float inv = 1.f / s;
        for (int c = 0; c < N_TOK; ++c) arow[c] *= inv;
        for (int c = N_TOK; c < NP; ++c) arow[c] = 0.f;
      } else {
        for (int c = 0; c < NP; ++c) arow[c] = 0.f;
      }
    }
    __syncthreads();

    // ===== GEMM3: out_h = P @ v, fuse 3x3 depthwise LePE =================
    for (int t = wave; t < 16; t += 8) {
      const int mt = t >> 2, nt = t & 3;
      v8f acc = (v8f){};
      const float* arow = attns + (mt * 16 + laneM) * NP;
      const int   d    = nt * 16 + laneM;                 // 0..63 (>=56 pad)
      const int   dc   = (d < HD) ? d : (HD - 1);
      const float dval = (d < HD) ? 1.f : 0.f;
      const float* vcol = qkvs + 2 * HD + dc;
      for (int kk = 0; kk < NP / 4; ++kk) {
        const int k0 = kk * 4 + laneH * 2;
        v2f a, bb;
        a.x  = arow[k0];
        a.y  = arow[k0 + 1];
        bb.x = vcol[k0 * HSLP] * dval;
        bb.y = vcol[(k0 + 1) * HSLP] * dval;
        acc = wmma_f32(a, bb, acc);
      }
      const int c = h * HD + dc;
      #pragma unroll
      for (int i = 0; i < 8; ++i) {
        const int nTok = mt * 16 + i + laneH * 8;
        float v = 0.f;
        if (nTok < N_TOK && d < HD) {
          const int i0 = nTok / 7, j0 = nTok % 7;
          float s = lepe_b[c];
          #pragma unroll
          for (int ky = 0; ky < 3; ++ky) {
            const int yi = i0 + ky - 1;
            #pragma unroll
            for (int kx = 0; kx < 3; ++kx) {
              const int xj = j0 + kx - 1;
              if (yi >= 0 && yi < 7 && xj >= 0 && xj < 7)
                s += qkvs[(yi * 7 + xj) * HSLP + 2 * HD + dc] *
                     lepe_w[c * 9 + ky * 3 + kx];
            }
          }
          v = acc[i] + s;
        }
        oh[nTok * NP + d] = v;  // padded rows/cols zeroed for GEMM4 K-pad
      }
    }
    __syncthreads();

    // ===== GEMM4: facc += out_h @ proj_w[h*56 : +56, :] ==================
    // All 6 tiles of this wave share mt: load A fragment once per K-step.
    {
      const float* arow = oh + (fmt * 16 + laneM) * NP;
      const float* pbase = proj_w + (size_t)h * HD * DIM;
      for (int kk = 0; kk < NP / 4; ++kk) {
        const int k0 = kk * 4 + laneH * 2;
        const float g0 = (k0 < HD) ? 1.f : 0.f;        // HD=56 < NP=64 pad
        const float g1 = (k0 + 1 < HD) ? 1.f : 0.f;
        const int k0c = (k0 < HD) ? k0 : 0;
        const int k1c = (k0 + 1 < HD) ? (k0 + 1) : 0;
        v2f a;
        a.x = arow[k0];
        a.y = arow[k0 + 1];
        #pragma unroll
        for (int j = 0; j < 6; ++j) {
          const int n = (fntb + j) * 16 + laneM;
          v2f bb;
          bb.x = pbase[k0c * DIM + n] * g0;
          bb.y = pbase[k1c * DIM + n] * g1;
          facc[j] = wmma_f32(a, bb, facc[j]);
        }
      }
    }
    __syncthreads();   // protect qkvs/attns/oh before next head rewrites
  }

  // ===== epilogue: + proj_b, single HBM write ============================
  #pragma unroll
  for (int j = 0; j < 6; ++j) {
    const int col = (fntb + j) * 16 + laneM;
    const float pb = proj_b[col];
    #pragma unroll
    for (int i = 0; i < 8; ++i) {
      const int nTok = fmt * 16 + i + laneH * 8;
      if (nTok < N_TOK)
        out[(size_t)b * N_TOK * DIM + nTok * DIM + col] = facc[j][i] + pb;
    }
  }
}

extern "C" void kernel_launch(void* const* d_in, const int* in_sizes, int n_in,
                              void* d_out, int out_size, void* d_ws, size_t ws_size,
                              hipStream_t stream) {
  const float* x         = (const float*)d_in[0];
  const float* mask      = (const float*)d_in[1];
  const float* qkv_w     = (const float*)d_in[2];
  const float* qkv_b     = (const float*)d_in[3];
  const float* bias_tab  = (const float*)d_in[4];
  const int*   rel_index = (const int*)  d_in[5];
  const float* lepe_w    = (const float*)d_in[6];
  const float* lepe_b    = (const float*)d_in[7];
  const float* proj_w    = (const float*)d_in[8];
  const float* proj_b    = (const float*)d_in[9];
  float* out = (float*)d_out;

  dim3 grid(2048);
  dim3 block(256);
  cswin_window_attn_kernel<<<grid, block, 0, stream>>>(
      x, mask, qkv_w, qkv_b, bias_tab, rel_index,
      lepe_w, lepe_b, proj_w, proj_b, out);
}